// Rescal_62405874811503
// MI455X (gfx1250) — compile-verified
//
#include <hip/hip_runtime.h>

typedef __attribute__((ext_vector_type(2))) float v2f;
typedef __attribute__((ext_vector_type(8))) float v8f;

#define DIMS   128
#define BATCH  16384
#define MARGIN 1.0f

// One wave32 per batch element.
// G(128x16) = R(128x128) @ T(128x16), T col0 = tail, col1 = c_tail, cols 2-15 = 0,
// via V_WMMA_F32_16X16X4_F32 (fp32 in, fp32 acc). Then pos = h.G[:,0],
// neg = ch.G[:,1], atomicAdd(relu(pos-neg+margin)/BATCH) into the scalar output.
__global__ void __launch_bounds__(256)
rescal_wmma_kernel(const int*   __restrict__ data,
                   const float* __restrict__ entities,
                   const float* __restrict__ relations,
                   float*       __restrict__ out)
{
    const int wavesPerBlock = blockDim.x >> 5;
    const int wave = blockIdx.x * wavesPerBlock + (threadIdx.x >> 5);
    const int lane = threadIdx.x & 31;

    // data layout per row: [head, tail, rel, c_head, c_tail]
    const int h_idx  = data[wave * 5 + 0];
    const int t_idx  = data[wave * 5 + 1];
    const int r_idx  = data[wave * 5 + 2];
    const int ch_idx = data[wave * 5 + 3];
    const int ct_idx = data[wave * 5 + 4];

    const float* __restrict__ R  = relations + (size_t)r_idx * (DIMS * DIMS);
    const float* __restrict__ h  = entities + (size_t)h_idx  * DIMS;
    const float* __restrict__ t  = entities + (size_t)t_idx  * DIMS;
    const float* __restrict__ ch = entities + (size_t)ch_idx * DIMS;
    const float* __restrict__ ct = entities + (size_t)ct_idx * DIMS;

    const int row0  = lane & 15;       // M (for A) / N (for B) within a tile
    const int khalf = (lane >> 4) * 2; // K sub-offset: lanes 0-15 -> {0,1}, 16-31 -> {2,3}

    // 8 accumulator tiles: G_m (16x16 f32), m = 0..7 covers M = 0..127
    v8f acc[8];
#pragma unroll
    for (int m = 0; m < 8; ++m) acc[m] = v8f{};

    // Branchless B source: N==0 -> tail, N==1 -> c_tail, N>=2 -> load tail's
    // address too (coalesces with lane 0's request) and zero via select.
    const float* __restrict__ bsrc = (row0 == 1) ? ct : t;
    const bool bactive = (row0 < 2);

#pragma unroll 4
    for (int ks = 0; ks < 32; ++ks) {   // K-steps of 4 over DIMS=128
        const int kbase = ks * 4 + khalf;

        // B operand (4x16 fp32, 2 VGPRs): unconditional load + cndmask zeroing.
        v2f braw = *(const v2f*)(bsrc + kbase);
        v2f b;
        b.x = bactive ? braw.x : 0.0f;
        b.y = bactive ? braw.y : 0.0f;

        // A operands: R tile rows m*16 + row0, cols kbase..kbase+1 (aligned float2)
        v2f a[8];
#pragma unroll
        for (int m = 0; m < 8; ++m) {
            a[m] = *(const v2f*)(R + (size_t)(m * 16 + row0) * DIMS + kbase);
        }

#pragma unroll
        for (int m = 0; m < 8; ++m) {
            acc[m] = __builtin_amdgcn_wmma_f32_16x16x4_f32(
                /*neg_a=*/false, a[m],
                /*neg_b=*/false, b,
                /*c_mod=*/(short)0, acc[m],
                /*reuse_a=*/false, /*reuse_b=*/false);
        }
    }

    // C/D layout: lane L holds column N = L&15; VGPR v holds M = v (lanes 0-15)
    // or M = v+8 (lanes 16-31). Column 0 -> pos (weight h), column 1 -> neg (weight ch).
    // Branchless epilogue: all lanes read valid weights, inactive lanes zeroed.
    const float* __restrict__ w = (row0 == 1) ? ch : h;
    const float scale = bactive ? 1.0f : 0.0f;
    const int moff = (lane >> 4) * 8;
    float partial = 0.0f;
#pragma unroll
    for (int m = 0; m < 8; ++m) {
#pragma unroll
        for (int v = 0; v < 8; ++v) {
            partial += w[m * 16 + moff + v] * acc[m][v];
        }
    }
    partial *= scale;

    const float pos = __shfl(partial, 0, 32) + __shfl(partial, 16, 32);
    const float neg = __shfl(partial, 1, 32) + __shfl(partial, 17, 32);
    const float loss = pos - neg + MARGIN;

    if (lane == 0 && loss > 0.0f) {
        atomicAdd(out, loss * (1.0f / (float)BATCH));
    }
}

__global__ void zero_out_kernel(float* out)
{
    if (threadIdx.x == 0) out[0] = 0.0f;
}

extern "C" void kernel_launch(void* const* d_in, const int* in_sizes, int n_in,
                              void* d_out, int out_size, void* d_ws, size_t ws_size,
                              hipStream_t stream) {
    (void)in_sizes; (void)n_in; (void)out_size; (void)d_ws; (void)ws_size;

    const int*   data      = (const int*)  d_in[0]; // (BATCH, 5) indices
    const float* entities  = (const float*)d_in[1]; // (500000, 128) f32
    const float* relations = (const float*)d_in[2]; // (200, 128, 128) f32
    float*       out       = (float*)d_out;         // scalar f32

    zero_out_kernel<<<1, 32, 0, stream>>>(out);

    const int wavesPerBlock = 8;                 // 256 threads = 8 wave32
    dim3 block(32 * wavesPerBlock);
    dim3 grid(BATCH / wavesPerBlock);            // 2048 blocks, exactly 16384 waves
    rescal_wmma_kernel<<<grid, block, 0, stream>>>(data, entities, relations, out);
}